// GELU179_39857296507268
// MI455X (gfx1250) — compile-verified
//
#include <hip/hip_runtime.h>
#include <math.h>

#define EPS      1e-5f
#define EPS_VAR  1e-4f
#define SQ2PI    0.7978845608028654f
#define GCOEF    0.044715f

#define BB   8
#define TT   4096
#define DD   1024
#define TC   64
#define NC   (TT/TC)          // 64 chunks

// workspace layout (in floats)
#define WS_SC      0                       // 16 scalars
#define WS_INVSTD  16                      // DD
#define WS_S       (16 + DD)               // BB*NC*DD  (chunk sums -> exclusive prefixes of x)
#define WS_SQ      (WS_S + BB*NC*DD)       // BB*NC*DD  (same for x^2)

typedef float v8f __attribute__((ext_vector_type(8)));
typedef float v2f __attribute__((ext_vector_type(2)));
typedef int   v4i __attribute__((ext_vector_type(4)));

// CDNA5 async global->LDS path (guarded; falls back to direct b128 global loads)
#if defined(__HIP_DEVICE_COMPILE__) && __has_builtin(__builtin_amdgcn_global_load_async_to_lds_b128)
#define HAS_ASYNC 1
typedef __attribute__((address_space(1))) v4i* gv4p;   // global int4*
typedef __attribute__((address_space(3))) v4i* lv4p;   // LDS int4*
#else
#define HAS_ASYNC 0
#endif

#if defined(__HIP_DEVICE_COMPILE__) && __has_builtin(__builtin_amdgcn_s_wait_asynccnt)
#define WAIT_ASYNC(n) __builtin_amdgcn_s_wait_asynccnt(n)
#else
#define WAIT_ASYNC(n) asm volatile("s_wait_asynccnt %0" :: "n"(n) : "memory")
#endif

__device__ __forceinline__ float spf(float v) {            // softplus, stable
    return (v > 20.0f) ? v : log1pf(expf(v));
}

// ---------------- Pass 1: scalars + per-d invstd ----------------
__global__ void __launch_bounds__(256) prep_kernel(
    const float* __restrict__ ema_mean, const float* __restrict__ ema_sq,
    const float* __restrict__ ema_out,  const float* __restrict__ var_fast,
    const float* __restrict__ var_slow,
    const float* lt, const float* ls1, const float* ls2, const float* ls3,
    const float* lw, const float* la1, const float* la2, const float* la3,
    float* __restrict__ ws)
{
    __shared__ float rbr[8], re2[8];
    int tid = threadIdx.x, lane = tid & 31, wave = tid >> 5;
    float br = 0.0f, e2 = 0.0f;
    for (int j = 0; j < 4; ++j) {
        int d = tid + 256 * j;
        float m  = ema_mean[d];
        float vg = fmaxf(ema_sq[d] - m * m, EPS_VAR);
        ws[WS_INVSTD + d] = 1.0f / (sqrtf(vg) + EPS);
        br += fminf(var_fast[d] / fmaxf(var_slow[d], EPS_VAR), 10.0f);
        float e = ema_out[d];
        e2 += e * e;
    }
    for (int m = 16; m >= 1; m >>= 1) {
        br += __shfl_xor(br, m, 32);
        e2 += __shfl_xor(e2, m, 32);
    }
    if (lane == 0) { rbr[wave] = br; re2[wave] = e2; }
    __syncthreads();
    if (tid == 0) {
        float brs = 0.0f, e2s = 0.0f;
        for (int k = 0; k < 8; ++k) { brs += rbr[k]; e2s += re2[k]; }
        float tau  = expf(lt[0]);
        float sig1 = spf(ls1[0]), sig2 = spf(ls2[0]), sig3 = spf(ls3[0]);
        float w    = spf(lw[0]);
        float a1   = spf(la1[0]), a2 = spf(la2[0]), a3 = spf(la3[0]);
        float burst = fmaxf(brs / (float)DD - 1.0f, 0.0f);
        float surp2 = tanhf(sig2 * burst);
        float s2p   = powf(fmaxf(surp2, 1e-7f), a2);
        float ne    = fmaxf(sqrtf(e2s), 1e-12f);
        ws[WS_SC + 0] = tau;  ws[WS_SC + 1] = sig1; ws[WS_SC + 2] = sig3;
        ws[WS_SC + 3] = w;    ws[WS_SC + 4] = a1;   ws[WS_SC + 5] = a3;
        ws[WS_SC + 6] = s2p;  ws[WS_SC + 7] = 1.0f / ne;
    }
}

// ---------------- Pass 2: chunk column-sums via V_WMMA_F32_16X16X4_F32 ----------------
// A = ones(16x4); B = x tile (K=4 t-rows x N=16 d-cols)  =>  D[m][n] = chunk column sum.
// Exact in f32; any K-row permutation of B leaves the column sums unchanged.
__global__ void __launch_bounds__(128) chunksum_wmma_kernel(
    const float* __restrict__ x, float* __restrict__ ws)
{
    int blk   = blockIdx.x;
    int ndblk = DD / 64;                 // 16 d-blocks of 64
    int dblk  = blk % ndblk;
    int bc    = blk / ndblk;
    int c     = bc % NC;
    int b     = bc / NC;

    int wave = threadIdx.x >> 5;
    int lane = threadIdx.x & 31;
    int half = lane >> 4;                // 0 or 1
    int d    = dblk * 64 + wave * 16 + (lane & 15);

    long base = ((long)b * TT + (long)c * TC) * DD + d;

    v2f a; a.x = 1.0f; a.y = 1.0f;       // all-ones A (16x4)
    v8f cs = {0.f,0.f,0.f,0.f,0.f,0.f,0.f,0.f};
    v8f cq = {0.f,0.f,0.f,0.f,0.f,0.f,0.f,0.f};

    for (int s = 0; s < TC / 4; ++s) {
        int t0 = 4 * s + 2 * half;       // low lanes: t {0,1}; high lanes: t {2,3}
        float x0 = x[base + (long)t0 * DD];
        float x1 = x[base + (long)(t0 + 1) * DD];
        v2f bx; bx.x = x0;      bx.y = x1;
        v2f bq; bq.x = x0 * x0; bq.y = x1 * x1;
        cs = __builtin_amdgcn_wmma_f32_16x16x4_f32(false, a, false, bx, (short)0, cs, false, false);
        cq = __builtin_amdgcn_wmma_f32_16x16x4_f32(false, a, false, bq, (short)0, cq, false, false);
    }

    // Every D row holds the same column sums; lanes n and n+16 both map to column n.
    long sidx = ((long)b * NC + c) * DD + d;
    if (half == 0) ws[WS_S  + sidx] = cs[0];
    else           ws[WS_SQ + sidx] = cq[0];
}

// ---------------- Pass 3: exclusive scan of chunk sums over c ----------------
__global__ void __launch_bounds__(256) scan_chunks_kernel(float* __restrict__ ws)
{
    int idx = blockIdx.x * blockDim.x + threadIdx.x;   // 0 .. BB*DD-1
    int b = idx / DD, d = idx % DD;
    float rs = 0.0f, rq = 0.0f;
    for (int c = 0; c < NC; ++c) {
        long i = ((long)b * NC + c) * DD + d;
        float s = ws[WS_S + i], q = ws[WS_SQ + i];
        ws[WS_S + i] = rs;  ws[WS_SQ + i] = rq;
        rs += s; rq += q;
    }
}

// ---------------- Pass 4: fused causal scan + row reductions + gate + output ----------------
// One block per (b, chunk); 256 threads x 4 contiguous d's cover all of D.
// Next x row is double-buffered into LDS with async b128 loads (ASYNCcnt pipeline);
// every thread reads back exactly the 16B it requested, so s_wait_asynccnt suffices.
__global__ void __launch_bounds__(256) scan_gate_kernel(
    const float* __restrict__ x, const float* __restrict__ ema_mean,
    const float* __restrict__ ema_out, const float* __restrict__ ws,
    float* __restrict__ out)
{
    __shared__ float red[4][8];
    __shared__ float gateSh;
    __shared__ float xbuf[2][DD];          // 8KB double buffer for async row staging

    int c = blockIdx.x % NC;
    int b = blockIdx.x / NC;
    int tid = threadIdx.x, lane = tid & 31, wave = tid >> 5;
    int d0 = tid * 4;

    float tau  = ws[WS_SC + 0], sig1 = ws[WS_SC + 1], sig3 = ws[WS_SC + 2];
    float w    = ws[WS_SC + 3], a1   = ws[WS_SC + 4], a3   = ws[WS_SC + 5];
    float s2p  = ws[WS_SC + 6], inv_ne = ws[WS_SC + 7];

    const float4 mean4 = ((const float4*)ema_mean)[tid];
    const float4 eo4   = ((const float4*)ema_out)[tid];
    const float4 is4   = ((const float4*)(ws + WS_INVSTD))[tid];
    long pb = ((long)b * NC + c) * (DD / 4);
    float4 cx4 = ((const float4*)(ws + WS_S))[pb + tid];   // exclusive prefix of x
    float4 cq4 = ((const float4*)(ws + WS_SQ))[pb + tid];  // exclusive prefix of x^2

    float cx[4] = {cx4.x, cx4.y, cx4.z, cx4.w};
    float cq[4] = {cq4.x, cq4.y, cq4.z, cq4.w};
    float mean_[4] = {mean4.x, mean4.y, mean4.z, mean4.w};
    float is_[4]   = {is4.x, is4.y, is4.z, is4.w};
    float eo_[4]   = {eo4.x, eo4.y, eo4.z, eo4.w};

    int t0 = c * TC;

#if HAS_ASYNC
    // prologue: stage first row of the chunk
    __builtin_amdgcn_global_load_async_to_lds_b128(
        (gv4p)(x + ((long)b * TT + t0) * DD + d0),
        (lv4p)&xbuf[0][d0], 0, 0);
#endif

    for (int tt = 0; tt < TC; ++tt) {
        int t = t0 + tt;
        long rowbase = ((long)b * TT + t) * DD;
        int cur = tt & 1;

        float xv[4];
#if HAS_ASYNC
        if (tt + 1 < TC) {
            __builtin_amdgcn_global_load_async_to_lds_b128(
                (gv4p)(x + (rowbase + DD) + d0),
                (lv4p)&xbuf[cur ^ 1][d0], 0, 0);
            WAIT_ASYNC(1);                 // current row's async load has landed
        } else {
            WAIT_ASYNC(0);
        }
        asm volatile("" ::: "memory");     // keep LDS reads after the wait
        const float4 xv4 = *(const float4*)&xbuf[cur][d0];
#else
        if (tt + 1 < TC)
            __builtin_prefetch(&x[rowbase + DD + d0], 0, 0);
        const float4 xv4 = ((const float4*)x)[rowbase / 4 + tid];
#endif
        xv[0] = xv4.x; xv[1] = xv4.y; xv[2] = xv4.z; xv[3] = xv4.w;

        float g[4];
        float p1 = 0.0f, p3 = 0.0f, pn = 0.0f, pd = 0.0f;
        float icnt = (t > 0) ? 1.0f / (float)t : 0.0f;   // cnt = clip(t,1); t==0 row zeroed

        for (int j = 0; j < 4; ++j) {
            float xi = xv[j];
            float u  = SQ2PI * (xi + GCOEF * xi * xi * xi);
            float gg = 0.5f * xi * (1.0f + tanhf(u));
            g[j] = gg;
            p1 += fabsf(xi - mean_[j]) * is_[j];
            if (t > 0) {
                float mu  = cx[j] * icnt;
                float sq  = cq[j] * icnt;
                float var = fmaxf(sq - mu * mu, EPS_VAR);
                p3 += fabsf(xi - mu) / (sqrtf(var) + EPS);
            }
            pn += gg * gg;
            pd += gg * eo_[j];
        }

        // wave32 reduce, then cross-wave via LDS
        for (int m = 16; m >= 1; m >>= 1) {
            p1 += __shfl_xor(p1, m, 32);
            p3 += __shfl_xor(p3, m, 32);
            pn += __shfl_xor(pn, m, 32);
            pd += __shfl_xor(pd, m, 32);
        }
        if (lane == 0) { red[0][wave] = p1; red[1][wave] = p3; red[2][wave] = pn; red[3][wave] = pd; }
        __syncthreads();
        if (tid == 0) {
            float s1s = 0, s3s = 0, n2 = 0, dt = 0;
            for (int k = 0; k < 8; ++k) { s1s += red[0][k]; s3s += red[1][k]; n2 += red[2][k]; dt += red[3][k]; }
            float surp1 = tanhf(sig1 * s1s / (float)DD);
            float surp3 = tanhf(sig3 * s3s / (float)DD);
            float s1 = powf(fmaxf(surp1, 1e-7f), a1);
            float s3 = powf(fmaxf(surp3, 1e-7f), a3);
            float joint = s1 * s2p * s3;
            float no   = fmaxf(sqrtf(n2), 1e-12f);
            float cosv = fminf(fmaxf(dt / no * inv_ne, -1.0f), 1.0f);
            gateSh = expf(-tau * cosv) * (1.0f + w * joint);
        }
        __syncthreads();
        float gate = gateSh;

        float4 o4;
        o4.x = g[0] * gate; o4.y = g[1] * gate; o4.z = g[2] * gate; o4.w = g[3] * gate;
        ((float4*)out)[rowbase / 4 + tid] = o4;
        for (int j = 0; j < 4; ++j) {
            cx[j] += xv[j];
            cq[j] += xv[j] * xv[j];
        }
        // red/gateSh reuse is safe: writers only pass the next iteration's first
        // barrier after all readers of this iteration have arrived there.
    }
}

extern "C" void kernel_launch(void* const* d_in, const int* in_sizes, int n_in,
                              void* d_out, int out_size, void* d_ws, size_t ws_size,
                              hipStream_t stream)
{
    const float* x        = (const float*)d_in[0];
    const float* ema_mean = (const float*)d_in[1];
    const float* ema_sq   = (const float*)d_in[2];
    const float* ema_out  = (const float*)d_in[3];
    const float* var_fast = (const float*)d_in[4];
    const float* var_slow = (const float*)d_in[5];
    const float* lt  = (const float*)d_in[6];
    const float* ls1 = (const float*)d_in[7];
    const float* ls2 = (const float*)d_in[8];
    const float* ls3 = (const float*)d_in[9];
    const float* lw  = (const float*)d_in[10];
    const float* la1 = (const float*)d_in[11];
    const float* la2 = (const float*)d_in[12];
    const float* la3 = (const float*)d_in[13];
    float* out = (float*)d_out;
    float* ws  = (float*)d_ws;

    prep_kernel<<<1, 256, 0, stream>>>(ema_mean, ema_sq, ema_out, var_fast, var_slow,
                                       lt, ls1, ls2, ls3, lw, la1, la2, la3, ws);

    chunksum_wmma_kernel<<<BB * NC * (DD / 64), 128, 0, stream>>>(x, ws);

    scan_chunks_kernel<<<(BB * DD) / 256, 256, 0, stream>>>(ws);

    scan_gate_kernel<<<BB * NC, 256, 0, stream>>>(x, ema_mean, ema_out, ws, out);
}